// AlignmentContrastiveLoss_51453708206435
// MI455X (gfx1250) — compile-verified
//
#include <hip/hip_runtime.h>

typedef __attribute__((ext_vector_type(2))) float v2f;
typedef __attribute__((ext_vector_type(4))) float v4f;
typedef __attribute__((ext_vector_type(8))) float v8f;

namespace {
constexpr int NB   = 128;   // batch
constexpr int L_IM = 50;
constexpr int L_S  = 40;
constexpr int DD   = 1024;  // feature dim (K)
constexpr int RN   = 49;    // valid im rows  (L_IM - 1)
constexpr int WN   = 37;    // valid s rows   (L_S - 3)
constexpr int RPAD = 64;    // padded rows  (4 x 16)
constexpr int WPAD = 48;    // padded cols  (3 x 16)
constexpr int KC   = 32;    // K chunk staged in LDS
constexpr int ASTR = 36;    // LDS row stride (dwords): 144B rows, 16B aligned, conflict free
constexpr int TSTR = 49;    // epilogue tile stride
constexpr int NT   = 384;   // 12 wave32s
constexpr int IMSZ = RPAD * ASTR;         // 2304 dwords
constexpr int SSZ  = WPAD * ASTR;         // 1728 dwords
constexpr int BUFSZ = IMSZ + SSZ;         // 4032 dwords of real data per buffer
constexpr int BUFSTRIDE = BUFSZ + 128;    // + per-buffer trash row (16B aligned)
constexpr int NCHUNK = DD / KC;           // 32
constexpr int NZROWS = (RPAD - RN) + (WPAD - WN);  // 15 + 11 = 26 padded rows per buffer
}

__global__ __launch_bounds__(NT)
void align_contrastive_kernel(const float* __restrict__ im_set,
                              const float* __restrict__ s_seq,
                              const int*   __restrict__ im_len,
                              const int*   __restrict__ s_len,
                              float*       __restrict__ out)
{
    // Double-buffered staging (2 x 16640 B); epilogue tile aliases buffer 0.
    __shared__ float smem[2 * BUFSTRIDE];

    const int j    = blockIdx.x;           // s batch index
    const int i    = blockIdx.y;           // im batch index
    const int tid  = threadIdx.x;
    const int lane = tid & 31;
    const int wv   = tid >> 5;             // 0..11
    const int rt   = wv / 3;               // 0..3 : 16-row tile
    const int wt   = wv % 3;               // 0..2 : 16-col tile

    const float* imBase = im_set + (size_t)i * (L_IM * DD) + DD;  // row r -> im_set[i, r+1]
    const float* sBase  = s_seq  + (size_t)j * (L_S * DD)  + DD;  // row w -> s_seq[j, w+1]

    const int il = im_len[i] - 1;
    const int sl = s_len[j]  - 3;

    // ---- one-time zero fill of padded LDS rows (both buffers) ----
    {
        const v4f z = {0.f, 0.f, 0.f, 0.f};
        for (int t = tid; t < 2 * NZROWS * (KC / 4); t += NT) {
            const int b  = t / (NZROWS * (KC / 4));
            const int q  = t % (NZROWS * (KC / 4));
            const int rr = q >> 3;                 // 0..25
            const int cc = (q & 7) << 2;
            const int rowoff = (rr < (RPAD - RN)) ? (RN + rr) * ASTR
                                                  : IMSZ + (WN + (rr - (RPAD - RN))) * ASTR;
            *(v4f*)&smem[b * BUFSTRIDE + rowoff + cc] = z;
        }
    }

    // ---- staging geometry ----
    // waves 0..7 stage im (slots: rows r0, r0+32); waves 8..11 stage s (rows r0, r0+16, r0+32).
    // ALL threads issue 3 loads per chunk (im role's 3rd = duplicate of slot 0, an L0 hit);
    // only the 3rd STORE is guarded by a wave-uniform scalar branch.
    const bool sRole = __builtin_amdgcn_readfirstlane(tid) >= 256;  // wave-uniform scalar
    const int  u     = sRole ? (tid - 256) : tid;
    const int  c4    = (u & 7) << 2;               // k offset within chunk (floats)
    const int  r0    = u >> 3;
    const int  r1    = r0 + (sRole ? 16 : 32);
    const int  r2    = r0 + 32;                    // s role only
    const int  lim   = sRole ? WN : RN;
    const bool v1    = r1 < lim;                   // s: always true; im: rows 32..48
    const bool v2    = r2 < WN;                    // s role slot 2

    const float* gbase = sRole ? sBase : imBase;
    const float* gp0 = gbase + r0 * DD + c4;
    const float* gp1 = gbase + (v1 ? r1 : 0) * DD + c4;             // clamped: always legal
    const float* gp2 = gbase + ((sRole && v2) ? r2 : r0) * DD + c4; // im role: dup of slot 0

    const int ldsbase = sRole ? IMSZ : 0;
    const int trash   = BUFSZ + lane * 4;          // per-buffer trash row
    const int lo0 = ldsbase + r0 * ASTR + c4;
    const int lo1 = v1 ? (ldsbase + r1 * ASTR + c4) : trash;
    const int lo2 = v2 ? (ldsbase + r2 * ASTR + c4) : trash;        // used only when sRole

    // ---- WMMA fragment offsets (dwords within a buffer) ----
    const int m  = lane & 15;
    const int kh = (lane >> 4) << 1;                       // K-half per ISA 16x4 f32 layout
    const int aoff = (rt * 16 + m) * ASTR + kh;            // A fragment base
    const int boff = IMSZ + (wt * 16 + m) * ASTR + kh;     // B fragment base

    v8f c = {};

    // ---- prologue: chunk 0 -> buffer 0 (visible at the first in-loop barrier) ----
    {
        v4f p0 = *(const v4f*)gp0;
        v4f p1 = *(const v4f*)gp1;
        v4f p2 = *(const v4f*)gp2;
        *(v4f*)&smem[lo0] = p0;
        *(v4f*)&smem[lo1] = p1;
        if (sRole) *(v4f*)&smem[lo2] = p2;
    }

    // ---- pipeline: prefetch lives entirely inside one barrier interval ----
    // barrier; issue loads for ch+1; WMMAs on ch (loads in flight); store ch+1 to buf^1.
    // By the next barrier's counter drain, loadcnt is already 0 -> no exposed latency.
    #pragma unroll
    for (int ch = 0; ch < NCHUNK; ++ch) {
        __syncthreads();
        float* buf  = smem + (ch & 1) * BUFSTRIDE;
        float* nbuf = smem + ((ch + 1) & 1) * BUFSTRIDE;
        const bool hasNext = (ch + 1 < NCHUNK);    // constant-folds under full unroll

        v4f ra0, ra1, ra2;
        if (hasNext) {
            const int k0 = (ch + 1) * KC;
            ra0 = *(const v4f*)(gp0 + k0);
            ra1 = *(const v4f*)(gp1 + k0);
            ra2 = *(const v4f*)(gp2 + k0);
        }

        #pragma unroll
        for (int t4 = 0; t4 < KC / 4; ++t4) {
            v2f a = *(const v2f*)&buf[aoff + (t4 << 2)];
            v2f b = *(const v2f*)&buf[boff + (t4 << 2)];
            // D = A(16x4,f32) * B(4x16,f32) + C(16x16,f32)
            c = __builtin_amdgcn_wmma_f32_16x16x4_f32(
                    false, a, false, b, (short)0, c, false, false);
        }

        if (hasNext) {
            *(v4f*)&nbuf[lo0] = ra0;
            *(v4f*)&nbuf[lo1] = ra1;
            if (sRole) *(v4f*)&nbuf[lo2] = ra2;    // uniform scalar branch
        }
    }

    // ---- epilogue: masked tile to LDS (aliases pipeline buffer 0) ----
    float* T = smem;                       // 64 x 49 (stride TSTR), 3136 dw < BUFSZ
    {
        const int w = wt * 16 + m;
        #pragma unroll
        for (int g = 0; g < 8; ++g) {
            const int r = rt * 16 + g + ((lane >> 4) << 3);
            float v = c[g];
            if (r >= il || w >= sl) v = 0.0f;   // reference mask (covers padding too)
            T[r * TSTR + w] = v;
        }
    }
    float* colmax = smem + RPAD * TSTR;    // 48 floats
    float* rowmax = colmax + WPAD;         // 64 floats
    __syncthreads();

    if (tid < WN) {
        float mx = -__builtin_inff();
        for (int r = 0; r < RN; ++r) mx = fmaxf(mx, T[r * TSTR + tid]);
        colmax[tid] = mx;
    } else if (tid >= 64 && tid < 64 + RN) {
        const int r = tid - 64;
        float mx = -__builtin_inff();
        for (int w = 0; w < WN; ++w) mx = fmaxf(mx, T[r * TSTR + w]);
        rowmax[r] = mx;
    }
    __syncthreads();

    if (tid == 0) {
        // deterministic serial sum: max-over-r summed over w + max-over-w summed over r
        float s = 0.f;
        for (int w = 0; w < WN; ++w) s += colmax[w];
        for (int r = 0; r < RN; ++r) s += rowmax[r];
        out[i * NB + j] = s;
    }
}

extern "C" void kernel_launch(void* const* d_in, const int* in_sizes, int n_in,
                              void* d_out, int out_size, void* d_ws, size_t ws_size,
                              hipStream_t stream) {
    (void)in_sizes; (void)n_in; (void)out_size; (void)d_ws; (void)ws_size;
    const float* im_set = (const float*)d_in[0];
    const float* s_seq  = (const float*)d_in[1];
    const int*   im_len = (const int*)d_in[2];
    const int*   s_len  = (const int*)d_in[3];
    float* out = (float*)d_out;

    dim3 grid(NB, NB);   // x = j (s batch), y = i (im batch): consecutive blocks reuse im[i] in L2
    align_contrastive_kernel<<<grid, NT, 0, stream>>>(im_set, s_seq, im_len, s_len, out);
}